// RLSLayer_40862318854347
// MI455X (gfx1250) — compile-verified
//
#include <hip/hip_runtime.h>

typedef float v2f __attribute__((ext_vector_type(2)));
typedef float v4f __attribute__((ext_vector_type(4)));
typedef float v8f __attribute__((ext_vector_type(8)));
typedef int   v4i __attribute__((ext_vector_type(4)));

#define RLS_T 64
#define RLS_BLOCK 128   // 4 wave32 per workgroup, one workgroup per batch item

__global__ __launch_bounds__(RLS_BLOCK) void rls_update_kernel(
    const float* __restrict__ x_g,   // (B, 64)
    const float* __restrict__ d_g,   // (B,)
    const float* __restrict__ w_g,   // (B, 64)
    const float* __restrict__ P_g,   // (B, 64, 64)
    const float* __restrict__ ff_g,  // scalar
    float* __restrict__ wn_g,        // (B, 64)
    float* __restrict__ Pn_g,        // (B, 64, 64)
    float* __restrict__ yh_g)        // (B,)
{
    __shared__ __align__(16) float Plds[RLS_T * RLS_T]; // 16 KB: the P tile
    __shared__ __align__(16) float xl[RLS_T];
    __shared__ __align__(16) float wl[RLS_T];
    __shared__ __align__(16) float Pxl[RLS_T];
    __shared__ __align__(16) float gl[RLS_T];

    const int b    = blockIdx.x;
    const int tid  = threadIdx.x;
    const int lane = tid & 31;
    const int wave = tid >> 5;

    const float* Pb = P_g + (size_t)b * (RLS_T * RLS_T);

    // ---- Stage x, w_prev into LDS (tiny, sync) ----
    if (tid < RLS_T) {
        xl[tid] = x_g[(size_t)b * RLS_T + tid];
        wl[tid] = w_g[(size_t)b * RLS_T + tid];
    }

    // ---- Stage P (16 KB) into LDS via async-to-LDS (CDNA5 ASYNCcnt path) ----
#if __has_builtin(__builtin_amdgcn_global_load_async_to_lds_b128)
    {
        typedef __attribute__((address_space(1))) v4i g1v4i;
        typedef __attribute__((address_space(3))) v4i l3v4i;
        g1v4i* gsrc = (g1v4i*)Pb;       // global int4*
        l3v4i* ldst = (l3v4i*)Plds;     // lds int4*
        #pragma unroll
        for (int it = 0; it < (RLS_T * RLS_T) / (RLS_BLOCK * 4); ++it) {
            const int vec = it * RLS_BLOCK + tid;   // 16-byte vector index
            __builtin_amdgcn_global_load_async_to_lds_b128(gsrc + vec, ldst + vec, 0, 0);
        }
    }
  #if __has_builtin(__builtin_amdgcn_s_wait_asynccnt)
    __builtin_amdgcn_s_wait_asynccnt(0);
  #else
    asm volatile("s_wait_asynccnt 0x0" ::: "memory");
  #endif
#else
    #pragma unroll
    for (int it = 0; it < (RLS_T * RLS_T) / (RLS_BLOCK * 4); ++it) {
        const int idx = (it * RLS_BLOCK + tid) * 4;
        *(v4f*)&Plds[idx] = *(const v4f*)&Pb[idx];
    }
#endif
    __syncthreads();

    // ---- Px = P @ x via chained V_WMMA_F32_16X16X4_F32 ----
    // Wave `wave` computes rows [16*wave, 16*wave+16).
    // A (16x4 f32, 2 VGPRs): lanes 0-15 -> M=lane, K=0..1 ; lanes 16-31 -> M=lane-16, K=2..3
    // B (4x16 f32, 2 VGPRs): x broadcast across all 16 N columns, K split like A.
    // D (16x16 f32, 8 VGPRs): every column holds Px tile; read column via lanes 0 and 16.
    {
        const int r0 = wave * 16;
        const int m  = lane & 15;
        const int kh = lane >> 4;                         // 0: K=0,1  1: K=2,3
        const float* Arow = &Plds[(r0 + m) * RLS_T + 2 * kh];
        v8f c = {};
        #pragma unroll
        for (int k0 = 0; k0 < RLS_T; k0 += 4) {
            v2f a, bv;
            a.x  = Arow[k0];
            a.y  = Arow[k0 + 1];
            bv.x = xl[k0 + 2 * kh];
            bv.y = xl[k0 + 2 * kh + 1];
            c = __builtin_amdgcn_wmma_f32_16x16x4_f32(
                    false, a, false, bv, (short)0, c, false, false);
        }
        if (m == 0) {                       // lanes 0 and 16 hold N==0 column
            const int base = r0 + kh * 8;   // VGPR v: M = v (lane 0) / v+8 (lane 16)
            #pragma unroll
            for (int v = 0; v < 8; ++v) Pxl[base + v] = c[v];
        }
    }
    __syncthreads();

    // ---- Scalars (redundant per thread; uniform LDS broadcasts) ----
    float lam = ff_g[0];
    lam = fminf(fmaxf(lam, 1e-4f), 0.9999f);
    float xPx = 0.0f, ywx = 0.0f;
    #pragma unroll 8
    for (int j = 0; j < RLS_T; ++j) {
        xPx = fmaf(xl[j], Pxl[j], xPx);
        ywx = fmaf(xl[j], wl[j], ywx);
    }
    const float denom  = lam + xPx;
    const float err    = d_g[b] - ywx;
    const float invLam = 1.0f / lam;

    // ---- gain, w_next, y_hat ----
    if (tid < RLS_T) {
        const float g = Pxl[tid] / denom;
        gl[tid] = g;
        wn_g[(size_t)b * RLS_T + tid] = fmaf(g, err, wl[tid]);
    }
    if (tid == 0) yh_g[b] = ywx;
    __syncthreads();

    // ---- P_next = (P - g * Px^T) / lam  (P symmetric => xP == Px) ----
    float* Pout = Pn_g + (size_t)b * (RLS_T * RLS_T);
    #pragma unroll
    for (int it = 0; it < (RLS_T * RLS_T) / (RLS_BLOCK * 4); ++it) {
        const int idx = (it * RLS_BLOCK + tid) * 4;   // contiguous, coalesced b128
        const int i   = idx >> 6;
        const int j   = idx & 63;                     // multiple of 4
        const v4f p   = *(const v4f*)&Plds[idx];
        const v4f px  = *(const v4f*)&Pxl[j];
        const float gi = gl[i];
        v4f o;
        o.x = (p.x - gi * px.x) * invLam;
        o.y = (p.y - gi * px.y) * invLam;
        o.z = (p.z - gi * px.z) * invLam;
        o.w = (p.w - gi * px.w) * invLam;
        *(v4f*)&Pout[idx] = o;
    }
}

extern "C" void kernel_launch(void* const* d_in, const int* in_sizes, int n_in,
                              void* d_out, int out_size, void* d_ws, size_t ws_size,
                              hipStream_t stream) {
    const float* x  = (const float*)d_in[0];  // (B, 64)
    const float* d  = (const float*)d_in[1];  // (B,)
    const float* w  = (const float*)d_in[2];  // (B, 64)
    const float* P  = (const float*)d_in[3];  // (B, 64, 64)
    const float* ff = (const float*)d_in[4];  // scalar

    const int B = in_sizes[1];                // 16384

    float* out = (float*)d_out;               // concat: w_next | P_next | y_hat
    float* wn  = out;
    float* Pn  = wn + (size_t)B * RLS_T;
    float* yh  = Pn + (size_t)B * RLS_T * RLS_T;

    rls_update_kernel<<<B, RLS_BLOCK, 0, stream>>>(x, d, w, P, ff, wn, Pn, yh);
}